// LeakyMinPlus_70832600646272
// MI455X (gfx1250) — compile-verified
//
#include <hip/hip_runtime.h>
#include <cstdint>

// Tropical (min-plus) linear layer: out[b,o] = min_k( X[b,k] + W[o,k] )
// X: [1024,512] f32, W: [512,512] f32, out: [1024,512] f32.
// VALU-bound (no WMMA mapping for the (min,+) semiring). CDNA5 path:
// async global->LDS b128 copies (ASYNCcnt) double-buffered; wave-swizzled
// thread map (8 tx x 4 ty per wave32) so fragment ds_load_b128 reads are
// broadcast with at most 2-way bank conflicts, with only 4 async ops per
// thread per K-slice (low VGPR pressure: no spills, no s_set_vgpr_msb).

#define BATCH 1024
#define KDIM  512
#define OUTF  512

#define BM 64
#define BN 64
#define BK 32
#define BKP 36     // LDS row stride (floats): 16B aligned; 4*36=144 = 64*2+16
#define TM 4
#define TN 4
#define NTHREADS 256

#if defined(__has_builtin)
#  if __has_builtin(__builtin_amdgcn_global_load_async_to_lds_b128)
#    define HAS_ASYNC_B128 1
#  endif
#  if __has_builtin(__builtin_amdgcn_s_wait_asynccnt)
#    define HAS_WAIT_ASYNC 1
#  endif
#endif
#ifndef HAS_ASYNC_B128
#define HAS_ASYNC_B128 0
#endif
#ifndef HAS_WAIT_ASYNC
#define HAS_WAIT_ASYNC 0
#endif

// async ops issued per thread per K-slice (compile-time constant for the wait)
#define TILE_ASYNC_OPS (4 * HAS_ASYNC_B128)

typedef int v4i __attribute__((ext_vector_type(4)));
typedef __attribute__((address_space(1))) v4i* global_v4i_ptr;
typedef __attribute__((address_space(3))) v4i* lds_v4i_ptr;

__device__ __forceinline__ void copy16_g2l(const float* g, float* l) {
#if HAS_ASYNC_B128
    __builtin_amdgcn_global_load_async_to_lds_b128(
        (global_v4i_ptr)(const_cast<float*>(g)), (lds_v4i_ptr)(l), 0, 0);
#else
    *(float4*)l = *(const float4*)g;
#endif
}

__device__ __forceinline__ void wait_tile(bool next_inflight) {
#if HAS_WAIT_ASYNC && (TILE_ASYNC_OPS > 0)
    if (next_inflight) __builtin_amdgcn_s_wait_asynccnt(TILE_ASYNC_OPS);
    else               __builtin_amdgcn_s_wait_asynccnt(0);
#else
    (void)next_inflight;   // dscnt/loadcnt handled by __syncthreads()
#endif
}

__global__ __launch_bounds__(NTHREADS)
void minplus_kernel(const float* __restrict__ X, const float* __restrict__ W,
                    float* __restrict__ out) {
    __shared__ float Xs[2][BM * BKP];   // row-major: Xs[m][k]
    __shared__ float Ws[2][BN * BKP];   // row-major: Ws[n][k]

    const int tid = threadIdx.x;
    const int n0  = blockIdx.x * BN;
    const int m0  = blockIdx.y * BM;

    // Wave-swizzled 16x16 thread map: within one wave32 (tid bits 4..0 vary),
    // tx spans 8 values (bits {tid5, tid2..0}) and ty spans 4 (bits {hi, tid4..3}).
    // -> W fragment reads: 8 addresses (2-way bank conflict, 4-lane broadcast)
    // -> X fragment reads: 4 addresses (conflict-free, 8-lane broadcast)
    const int tx = (tid & 7) | ((tid >> 2) & 8);          // 0..15
    const int ty = ((tid >> 3) & 3) | ((tid >> 4) & 12);  // 0..15

    // Copy geometry: ONE 32-bit element offset per matrix; each async issue
    // adds only compile-time constants.
    const uint32_t crow = (uint32_t)tid >> 3;          // 0..31
    const uint32_t cc4  = ((uint32_t)tid & 7u) << 2;   // 0,4,...,28
    uint32_t goffX = ((uint32_t)m0 + crow) * KDIM + cc4;   // += BK per slice
    uint32_t goffW = ((uint32_t)n0 + crow) * KDIM + cc4;
    const uint32_t loff = crow * BKP + cc4;                // element offset in buffer

    auto issue_tile = [&](int buf) {
        float* xb = &Xs[buf][0];
        float* wb = &Ws[buf][0];
#pragma unroll
        for (uint32_t i = 0; i < 2; ++i) {
            copy16_g2l(&X[goffX + i * (32u * KDIM)], xb + (loff + i * (32u * BKP)));
            copy16_g2l(&W[goffW + i * (32u * KDIM)], wb + (loff + i * (32u * BKP)));
        }
        goffX += BK;                    // advance to next K-slice
        goffW += BK;
    };

    float acc[TM][TN];
#pragma unroll
    for (int r = 0; r < TM; ++r)
#pragma unroll
        for (int c = 0; c < TN; ++c) acc[r][c] = __builtin_inff();

    const int NT = KDIM / BK;   // 16 K-slices
    issue_tile(0);

#pragma unroll 1
    for (int t = 0; t < NT; ++t) {
        const int buf = t & 1;
        const bool more = (t + 1) < NT;
        if (more) issue_tile(buf ^ 1);   // prefetch next slice into other buffer
        wait_tile(more);                 // this wave's slice-t copies landed
        __syncthreads();                 // all waves' slice-t data visible

        const float* xp = &Xs[buf][(ty * TM) * BKP];
        const float* wp = &Ws[buf][(tx * TN) * BKP];

#pragma unroll
        for (int kk = 0; kk < BK; kk += 4) {
            float4 a[TM];       // a[r] = X[m=ty*4+r][kk..kk+3]
#pragma unroll
            for (int r = 0; r < TM; ++r)
                a[r] = *(const float4*)(xp + r * BKP + kk);
            float4 b[TN];       // b[c] = W[n=tx*4+c][kk..kk+3]
#pragma unroll
            for (int c = 0; c < TN; ++c)
                b[c] = *(const float4*)(wp + c * BKP + kk);
#pragma unroll
            for (int r = 0; r < TM; ++r) {
                const float ax = a[r].x, ay = a[r].y, az = a[r].z, aw = a[r].w;
#pragma unroll
                for (int c = 0; c < TN; ++c) {
                    float acv = acc[r][c];
                    // fminf(fminf(x,y),z) -> v_min3_num_f32 candidates
                    acv = fminf(fminf(acv, ax + b[c].x), ay + b[c].y);
                    acv = fminf(fminf(acv, az + b[c].z), aw + b[c].w);
                    acc[r][c] = acv;
                }
            }
        }
        if (more) __syncthreads();   // done reading buf before t+2 refills it
    }

    // float4 stores; per wave: 4 rows x 8 consecutive tx -> 128B segments
#pragma unroll
    for (int r = 0; r < TM; ++r) {
        float4 v;
        v.x = acc[r][0]; v.y = acc[r][1]; v.z = acc[r][2]; v.w = acc[r][3];
        *(float4*)&out[(uint32_t)(m0 + ty * TM + r) * OUTF + (uint32_t)(n0 + tx * TN)] = v;
    }
}

extern "C" void kernel_launch(void* const* d_in, const int* in_sizes, int n_in,
                              void* d_out, int out_size, void* d_ws, size_t ws_size,
                              hipStream_t stream) {
    (void)in_sizes; (void)n_in; (void)out_size; (void)d_ws; (void)ws_size;
    const float* X = (const float*)d_in[0];   // [1024, 512]
    const float* W = (const float*)d_in[1];   // [512, 512]
    float* out     = (float*)d_out;           // [1024, 512]

    dim3 grid(OUTF / BN, BATCH / BM);         // (8, 16)
    minplus_kernel<<<grid, NTHREADS, 0, stream>>>(X, W, out);
}